// CapsuleLayer_51410758533727
// MI455X (gfx1250) — compile-verified
//
#include <hip/hip_runtime.h>

// CapsuleLayer dynamic routing, fused for MI455X (gfx1250, wave32).
// hat is never materialized to HBM: it is recomputed per routing pass with
// V_WMMA_F32_16X16X4_F32 (per-n GEMM  Hat_n[64x512] = X_n[64x8] @ W_n^T[8x512]),
// keeping the working set (W 64MB + x 8MB + logits 32MB) L2-resident.

typedef float v2f __attribute__((ext_vector_type(2)));
typedef float v8f __attribute__((ext_vector_type(8)));

#define B_  64
#define N_  4096
#define I_  8
#define C_  32
#define D_  16

#define PASS_GRID 256
#define N_PER_WG  (N_ / PASS_GRID)   // 16 n per workgroup

__global__ void caps_zero_kernel(float* __restrict__ p, int n) {
    int i = blockIdx.x * blockDim.x + threadIdx.x;
    if (i < n) p[i] = 0.0f;
}

// squash(v) = (s2/(1+s2)) * v / sqrt(s2 + eps), over j (16 elements)
__global__ void caps_squash_kernel(const float* __restrict__ acc,
                                   float* __restrict__ dst) {
    int t = blockIdx.x * blockDim.x + threadIdx.x;   // one (b,c) pair per thread
    if (t >= B_ * C_) return;
    const float* v = acc + (size_t)t * D_;
    float tmp[D_];
    float s2 = 0.0f;
    #pragma unroll
    for (int j = 0; j < D_; ++j) { tmp[j] = v[j]; s2 += tmp[j] * tmp[j]; }
    float f = (s2 / (1.0f + s2)) * rsqrtf(s2 + 1e-7f);
    #pragma unroll
    for (int j = 0; j < D_; ++j) dst[(size_t)t * D_ + j] = tmp[j] * f;
}

// One routing pass. ITER==0: uniform coupling 1/32, pure WMMA accumulation.
// ITER==1: logits = hat.out_prev (b was 0), store b1 to blog, softmax, accumulate.
// ITER==2: logits = blog + hat.out_prev, softmax, accumulate.
template <int ITER>
__global__ __launch_bounds__(256, 1)
void caps_pass_kernel(const float* __restrict__ x,     // [B,N,I]
                      const float* __restrict__ W,     // [C,N,D,I]
                      const float* __restrict__ out_prev, // [B,C,D] (ITER>0)
                      float* __restrict__ blog,        // [B,C,N]
                      float* __restrict__ acc)         // [B,C,D] accumulator
{
    __shared__ float s_logit[B_ * C_];
    __shared__ float s_cw[B_ * C_];

    const int tid  = threadIdx.x;
    const int wave = tid >> 5;
    const int lane = tid & 31;
    const int b0 = (wave >> 1) * 16;      // batch tile base (4 tiles of 16)
    const int c0 = (wave & 1) * 16;       // capsule half (16 c per wave)
    const int jj = lane & 15;             // N index (j) / M index inside tile
    const int hi = lane >> 4;             // lane half
    const int i0 = hi * 2;                // K offset within K-group-of-4
    const int brow = hi * 8;              // +8 rows for high lane half (C/D layout)

    v8f po[16];                           // partial out[b-tile, c, j], 128 VGPRs
    #pragma unroll
    for (int c = 0; c < 16; ++c) po[c] = (v8f)0.0f;
    const v8f zero8 = (v8f)0.0f;

    const int n_base = blockIdx.x * N_PER_WG;

    for (int nn = 0; nn < N_PER_WG; ++nn) {
        const int n = n_base + nn;

        // A fragments (16x4 fp32, two K-halves): A[M,K] = x[b0+M, n, K]
        const float* xp = x + ((size_t)(b0 + jj) * N_ + n) * I_;
        v2f a0, a1;
        { float2 t0 = *(const float2*)(xp + i0);     a0[0] = t0.x; a0[1] = t0.y;
          float2 t1 = *(const float2*)(xp + i0 + 4); a1[0] = t1.x; a1[1] = t1.y; }

        if constexpr (ITER == 0) {
            // c = 1/32 uniform: po accumulates hat directly through WMMA C operand
            #pragma unroll
            for (int c = 0; c < 16; ++c) {
                const float* wp = W + ((size_t)(c0 + c) * N_ + n) * (D_ * I_) + jj * I_;
                v2f w0, w1;
                { float2 t0 = *(const float2*)(wp + i0);     w0[0] = t0.x; w0[1] = t0.y;
                  float2 t1 = *(const float2*)(wp + i0 + 4); w1[0] = t1.x; w1[1] = t1.y; }
                po[c] = __builtin_amdgcn_wmma_f32_16x16x4_f32(
                            false, a0, false, w0, (short)0, po[c], false, false);
                po[c] = __builtin_amdgcn_wmma_f32_16x16x4_f32(
                            false, a1, false, w1, (short)0, po[c], false, false);
            }
        } else {
            v8f h[16];                    // hat tiles, kept live across barriers
            // ---- phase 1: hat tiles + agreement logits -----------------------
            #pragma unroll
            for (int c = 0; c < 16; ++c) {
                const float* wp = W + ((size_t)(c0 + c) * N_ + n) * (D_ * I_) + jj * I_;
                v2f w0, w1;
                { float2 t0 = *(const float2*)(wp + i0);     w0[0] = t0.x; w0[1] = t0.y;
                  float2 t1 = *(const float2*)(wp + i0 + 4); w1[0] = t1.x; w1[1] = t1.y; }
                v8f hh = __builtin_amdgcn_wmma_f32_16x16x4_f32(
                             false, a0, false, w0, (short)0, zero8, false, false);
                hh = __builtin_amdgcn_wmma_f32_16x16x4_f32(
                             false, a1, false, w1, (short)0, hh, false, false);
                h[c] = hh;

                // dot over j with out_prev[b, c, :] (L2-hot, 128 KB total)
                const float* op = out_prev +
                    ((size_t)(b0 + brow) * C_ + (c0 + c)) * D_ + jj;
                #pragma unroll
                for (int r = 0; r < 8; ++r) {
                    float p = hh[r] * op[(size_t)r * C_ * D_];
                    p += __shfl_xor(p, 1, 32);
                    p += __shfl_xor(p, 2, 32);
                    p += __shfl_xor(p, 4, 32);
                    p += __shfl_xor(p, 8, 32);   // sum over 16 j's per lane-half
                    if (jj == 0) {
                        int b = b0 + r + brow;
                        float lg = p;
                        if constexpr (ITER == 2)
                            lg += blog[((size_t)b * C_ + (c0 + c)) * N_ + n];
                        s_logit[b * C_ + (c0 + c)] = lg;
                        if constexpr (ITER == 1)
                            blog[((size_t)b * C_ + (c0 + c)) * N_ + n] = lg;
                    }
                }
            }
            __syncthreads();
            // ---- phase 2: softmax over c per b -------------------------------
            if (tid < B_) {
                float m = -3.4e38f;
                for (int c = 0; c < C_; ++c)
                    m = fmaxf(m, s_logit[tid * C_ + c]);
                float s = 0.0f;
                for (int c = 0; c < C_; ++c)
                    s += __expf(s_logit[tid * C_ + c] - m);
                float inv = 1.0f / s;
                for (int c = 0; c < C_; ++c)
                    s_cw[tid * C_ + c] = __expf(s_logit[tid * C_ + c] - m) * inv;
            }
            __syncthreads();
            // ---- phase 3: po += c_w * hat ------------------------------------
            #pragma unroll
            for (int c = 0; c < 16; ++c) {
                #pragma unroll
                for (int r = 0; r < 8; ++r) {
                    float f = s_cw[(b0 + r + brow) * C_ + (c0 + c)];
                    po[c][r] += f * h[c][r];
                }
            }
        }
    }

    // flush partial out via global f32 atomics (acc is 128 KB, L2-resident)
    const float scale = (ITER == 0) ? (1.0f / 32.0f) : 1.0f;
    #pragma unroll
    for (int c = 0; c < 16; ++c) {
        #pragma unroll
        for (int r = 0; r < 8; ++r) {
            int b = b0 + r + brow;
            atomicAdd(&acc[((size_t)b * C_ + (c0 + c)) * D_ + jj], po[c][r] * scale);
        }
    }
}

extern "C" void kernel_launch(void* const* d_in, const int* in_sizes, int n_in,
                              void* d_out, int out_size, void* d_ws, size_t ws_size,
                              hipStream_t stream) {
    const float* x = (const float*)d_in[0];   // [64,4096,8]
    const float* W = (const float*)d_in[1];   // [32,4096,16,8]
    float* out = (float*)d_out;               // [64,32,16]
    float* ws  = (float*)d_ws;

    // workspace layout (floats): acc[3*32768] | out0[32768] | out1[32768] | blog[8388608]
    // total ~34.2 MB
    float* acc  = ws;
    float* out0 = ws + 3 * (B_ * C_ * D_);
    float* out1 = out0 + (B_ * C_ * D_);
    float* blog = out1 + (B_ * C_ * D_);

    const int accN = 3 * B_ * C_ * D_;        // 98304
    caps_zero_kernel<<<(accN + 255) / 256, 256, 0, stream>>>(acc, accN);

    float* acc0 = acc;
    float* acc1 = acc + (B_ * C_ * D_);
    float* acc2 = acc + 2 * (B_ * C_ * D_);

    // routing iteration 0 (uniform coupling)
    caps_pass_kernel<0><<<PASS_GRID, 256, 0, stream>>>(x, W, nullptr, blog, acc0);
    caps_squash_kernel<<<(B_ * C_ + 255) / 256, 256, 0, stream>>>(acc0, out0);
    // routing iteration 1 (b1 = out0 . hat, stored)
    caps_pass_kernel<1><<<PASS_GRID, 256, 0, stream>>>(x, W, out0, blog, acc1);
    caps_squash_kernel<<<(B_ * C_ + 255) / 256, 256, 0, stream>>>(acc1, out1);
    // routing iteration 2 (b2 = b1 + out1 . hat)
    caps_pass_kernel<2><<<PASS_GRID, 256, 0, stream>>>(x, W, out1, blog, acc2);
    caps_squash_kernel<<<(B_ * C_ + 255) / 256, 256, 0, stream>>>(acc2, out);
}